// S2GNNLayer_55628416418032
// MI455X (gfx1250) — compile-verified
//
#include <hip/hip_runtime.h>

// S2GNN layer for MI455X (gfx1250, wave32, WMMA).
// Dense math in full fp32 via v_wmma_f32_16x16x4_f32 (matches fp32 reference).
// LDS staging uses CDNA5 async global->LDS copies (ASYNCcnt path).
// Edge scatter is L2-atomic-bound (xw and out both fit in 192MB L2).

typedef __attribute__((ext_vector_type(2))) float v2f;
typedef __attribute__((ext_vector_type(8))) float v8f;

#define NNODES 100000
#define NEDGES 3200000
#define KEIG   64
#define CCH    128

__device__ __forceinline__ v8f wmma_f32(v2f a, v2f b, v8f c) {
  // (neg_a, A, neg_b, B, c_mod, C, reuse_a, reuse_b)
  return __builtin_amdgcn_wmma_f32_16x16x4_f32(false, a, false, b, (short)0, c,
                                               false, false);
}

// CDNA5 async copy: 16B per lane, memory -> LDS, no VGPR data path.
// lds_off = per-lane LDS byte offset (low 32 bits of a __shared__ pointer),
// gptr    = per-lane global address (VGPR pair).
__device__ __forceinline__ void async_copy_b128(unsigned lds_off, const void* gptr) {
  asm volatile("global_load_async_to_lds_b128 %0, %1, off"
               :: "v"(lds_off), "v"(gptr)
               : "memory");
}
__device__ __forceinline__ void wait_async0() {
  asm volatile("s_wait_asynccnt 0x0" ::: "memory");
}

// ---------------- degree / normalization ----------------
__global__ __launch_bounds__(256) void init_deg(float* __restrict__ deg) {
  int i = blockIdx.x * 256 + threadIdx.x;
  if (i < NNODES) deg[i] = 1.0f;  // self-loop contribution
}

__global__ __launch_bounds__(256) void count_deg(const long long* __restrict__ ei,
                                                 float* __restrict__ deg) {
  int e = blockIdx.x * 256 + threadIdx.x;
  if (e < NEDGES) {
    int d = (int)ei[(size_t)NEDGES + e];  // dst row
    atomicAdd(&deg[d], 1.0f);
  }
}

__global__ __launch_bounds__(256) void make_dinv(float* __restrict__ deg) {
  int i = blockIdx.x * 256 + threadIdx.x;
  if (i < NNODES) deg[i] = rsqrtf(deg[i]);  // deg >= 1 always
}

__global__ __launch_bounds__(256) void zero_buf(float* __restrict__ p, int n) {
  int i = blockIdx.x * 256 + threadIdx.x;
  if (i < n) p[i] = 0.0f;
}

// ---------------- GEMM1: XW = X(N,128) @ W(128,128) ----------------
// Block = 8 waves, 128 rows. W staged in LDS (64KB) via async global->LDS.
// Each wave: 16 rows x 8 column tiles, 32 k-steps of V_WMMA_F32_16X16X4_F32;
// A frag reused across the 8 column tiles.
__global__ __launch_bounds__(256) void gemm_xw(const float* __restrict__ X,
                                               const float* __restrict__ W,
                                               float* __restrict__ XW) {
  __shared__ float sW[CCH * CCH];
  const int tid = threadIdx.x;
  {
    const unsigned ldsW = (unsigned)(uintptr_t)&sW[0];
    const float4* Wv = (const float4*)W;
    for (int i = tid; i < CCH * CCH / 4; i += 256)
      async_copy_b128(ldsW + (unsigned)i * 16u, (const void*)(Wv + i));
    wait_async0();
  }
  __syncthreads();

  const int wave = tid >> 5;
  const int lane = tid & 31;
  const int half = lane >> 4;   // 0: K pair {0,1}, 1: K pair {2,3}
  const int mrow = lane & 15;
  const int r0 = blockIdx.x * 128 + wave * 16;

  int arow = r0 + mrow;
  if (arow >= NNODES) arow = NNODES - 1;  // clamp; stores are guarded
  const float* Xr = X + (size_t)arow * CCH;

  v8f acc[8] = {};
  for (int k0 = 0; k0 < CCH; k0 += 4) {
    float2 av = *(const float2*)(Xr + k0 + 2 * half);
    v2f a; a.x = av.x; a.y = av.y;
#pragma unroll
    for (int ct = 0; ct < 8; ++ct) {
      v2f b;
      b.x = sW[(k0 + 2 * half + 0) * CCH + ct * 16 + mrow];
      b.y = sW[(k0 + 2 * half + 1) * CCH + ct * 16 + mrow];
      acc[ct] = wmma_f32(a, b, acc[ct]);
    }
  }
#pragma unroll
  for (int ct = 0; ct < 8; ++ct) {
    const int col = ct * 16 + mrow;
#pragma unroll
    for (int v = 0; v < 8; ++v) {
      const int row = r0 + v + 8 * half;
      if (row < NNODES) XW[(size_t)row * CCH + col] = acc[ct][v];
    }
  }
}

// ---------------- projection: Xhat(64,128) += Vk_chunk^T @ X_chunk ----------
// Block reduces 1024 rows; 8 waves cover 4 M-tiles x 8 N-tiles (4 tiles/wave).
__global__ __launch_bounds__(256) void project_vk(const float* __restrict__ Vk,
                                                  const float* __restrict__ X,
                                                  float* __restrict__ Xhat) {
  const int tid = threadIdx.x, wave = tid >> 5, lane = tid & 31;
  const int half = lane >> 4, mrow = lane & 15;
  const int mt = wave >> 1;            // 0..3  (K-rows 16*mt..)
  const int ctBase = (wave & 1) * 4;   // 0 or 4
  const int nStart = blockIdx.x * 1024;

  v8f acc[4] = {};
  for (int n0 = nStart; n0 < nStart + 1024; n0 += 4) {
    const int na = n0 + 2 * half;  // this lane covers rows na, na+1
    v2f a;
    a.x = (na     < NNODES) ? Vk[(size_t)(na    ) * KEIG + mt * 16 + mrow] : 0.f;
    a.y = (na + 1 < NNODES) ? Vk[(size_t)(na + 1) * KEIG + mt * 16 + mrow] : 0.f;
#pragma unroll
    for (int t = 0; t < 4; ++t) {
      const int col = (ctBase + t) * 16 + mrow;
      v2f b;
      b.x = (na     < NNODES) ? X[(size_t)(na    ) * CCH + col] : 0.f;
      b.y = (na + 1 < NNODES) ? X[(size_t)(na + 1) * CCH + col] : 0.f;
      acc[t] = wmma_f32(a, b, acc[t]);
    }
  }
#pragma unroll
  for (int t = 0; t < 4; ++t) {
    const int col = (ctBase + t) * 16 + mrow;
#pragma unroll
    for (int v = 0; v < 8; ++v) {
      const int krow = mt * 16 + v + 8 * half;
      atomicAdd(&Xhat[krow * CCH + col], acc[t][v]);
    }
  }
}

// ---------------- mix: Y2(64,128) = (gain . Xhat) @ Wlin^T ----------------
// Single block; same 4-tiles-per-wave split. Both fragments are contiguous
// float2 loads (Wlin^T element (j,c) = Wlin[c*128 + j]).
__global__ __launch_bounds__(256) void mix_spectral(const float* __restrict__ Xhat,
                                                    const float* __restrict__ gain,
                                                    const float* __restrict__ Wlin,
                                                    float* __restrict__ Y2) {
  const int tid = threadIdx.x, wave = tid >> 5, lane = tid & 31;
  const int half = lane >> 4, mrow = lane & 15;
  const int mt = wave >> 1;
  const int ctBase = (wave & 1) * 4;

  const int arow = mt * 16 + mrow;  // K-index of this lane's A row
  const float g = gain[arow];

  v8f acc[4] = {};
  for (int j0 = 0; j0 < CCH; j0 += 4) {
    float2 av = *(const float2*)(Xhat + arow * CCH + j0 + 2 * half);
    v2f a; a.x = av.x * g; a.y = av.y * g;
#pragma unroll
    for (int t = 0; t < 4; ++t) {
      const int c = (ctBase + t) * 16 + mrow;  // output channel (B's N index)
      float2 bv = *(const float2*)(Wlin + (size_t)c * CCH + j0 + 2 * half);
      v2f b; b.x = bv.x; b.y = bv.y;
      acc[t] = wmma_f32(a, b, acc[t]);
    }
  }
#pragma unroll
  for (int t = 0; t < 4; ++t) {
    const int col = (ctBase + t) * 16 + mrow;
#pragma unroll
    for (int v = 0; v < 8; ++v) {
      const int krow = mt * 16 + v + 8 * half;
      Y2[krow * CCH + col] = acc[t][v];
    }
  }
}

// -------- lift + base: out = Vk @ Y2 + XW*dinv^2 (self loop) + b ----------
__global__ __launch_bounds__(256) void lift_base(const float* __restrict__ Vk,
                                                 const float* __restrict__ Y2,
                                                 const float* __restrict__ XW,
                                                 const float* __restrict__ dinv,
                                                 const float* __restrict__ bgcn,
                                                 float* __restrict__ out) {
  __shared__ float sY[KEIG * CCH];  // 32KB, staged via async global->LDS
  const int tid = threadIdx.x;
  {
    const unsigned ldsY = (unsigned)(uintptr_t)&sY[0];
    const float4* Yv = (const float4*)Y2;
    for (int i = tid; i < KEIG * CCH / 4; i += 256)
      async_copy_b128(ldsY + (unsigned)i * 16u, (const void*)(Yv + i));
    wait_async0();
  }
  __syncthreads();

  const int wave = tid >> 5, lane = tid & 31;
  const int half = lane >> 4, mrow = lane & 15;
  const int r0 = blockIdx.x * 128 + wave * 16;

  int arow = r0 + mrow;
  if (arow >= NNODES) arow = NNODES - 1;
  const float* Vr = Vk + (size_t)arow * KEIG;

  v8f acc[8] = {};
  for (int k0 = 0; k0 < KEIG; k0 += 4) {
    float2 av = *(const float2*)(Vr + k0 + 2 * half);
    v2f a; a.x = av.x; a.y = av.y;
#pragma unroll
    for (int ct = 0; ct < 8; ++ct) {
      v2f b;
      b.x = sY[(k0 + 2 * half + 0) * CCH + ct * 16 + mrow];
      b.y = sY[(k0 + 2 * half + 1) * CCH + ct * 16 + mrow];
      acc[ct] = wmma_f32(a, b, acc[ct]);
    }
  }
#pragma unroll
  for (int ct = 0; ct < 8; ++ct) {
    const int col = ct * 16 + mrow;
    const float bb = bgcn[col];
#pragma unroll
    for (int v = 0; v < 8; ++v) {
      const int row = r0 + v + 8 * half;
      if (row < NNODES) {
        const float di = dinv[row];
        out[(size_t)row * CCH + col] =
            acc[ct][v] + XW[(size_t)row * CCH + col] * di * di + bb;
      }
    }
  }
}

// ---------------- edge scatter: out[dst] += XW[src] * dinv[s]*dinv[d] -------
// One wave per edge: float4 gather, 4 atomics/lane (L2-resident working set).
__global__ __launch_bounds__(256) void edge_scatter(const long long* __restrict__ ei,
                                                    const float* __restrict__ XW,
                                                    const float* __restrict__ dinv,
                                                    float* __restrict__ out) {
  const int wave = threadIdx.x >> 5, lane = threadIdx.x & 31;
  const long long e = (long long)blockIdx.x * 8 + wave;
  if (e >= NEDGES) return;
  const long long s = ei[e];
  const long long d = ei[(size_t)NEDGES + e];
  const float norm = dinv[s] * dinv[d];
  const float4 v = ((const float4*)(XW + (size_t)s * CCH))[lane];
  float* dst = out + (size_t)d * CCH + lane * 4;
  atomicAdd(dst + 0, v.x * norm);
  atomicAdd(dst + 1, v.y * norm);
  atomicAdd(dst + 2, v.z * norm);
  atomicAdd(dst + 3, v.w * norm);
}

// ---------------- relu ----------------
__global__ __launch_bounds__(256) void relu_k(float* __restrict__ out) {
  const int i = blockIdx.x * 256 + threadIdx.x;
  if (i < NNODES * CCH / 4) {
    float4* p = (float4*)out;
    float4 v = p[i];
    v.x = fmaxf(v.x, 0.f); v.y = fmaxf(v.y, 0.f);
    v.z = fmaxf(v.z, 0.f); v.w = fmaxf(v.w, 0.f);
    p[i] = v;
  }
}

extern "C" void kernel_launch(void* const* d_in, const int* in_sizes, int n_in,
                              void* d_out, int out_size, void* d_ws, size_t ws_size,
                              hipStream_t stream) {
  (void)in_sizes; (void)n_in; (void)out_size; (void)ws_size;
  const float*     x     = (const float*)d_in[0];      // [N,128]
  const long long* ei    = (const long long*)d_in[1];  // [2,E] int64
  const float*     Vk    = (const float*)d_in[2];      // [N,64]
  const float*     gain  = (const float*)d_in[3];      // [64]
  const float*     W_gcn = (const float*)d_in[4];      // [128,128]
  const float*     b_gcn = (const float*)d_in[5];      // [128]
  const float*     W_lin = (const float*)d_in[6];      // [128,128]
  float* out = (float*)d_out;

  // Workspace layout (~51.7 MB of f32)
  float* XW   = (float*)d_ws;                      // N*128
  float* dinv = XW + (size_t)NNODES * CCH;         // N
  float* Xhat = dinv + NNODES;                     // 64*128
  float* Y2   = Xhat + KEIG * CCH;                 // 64*128

  // 1-3: symmetric normalization factors
  init_deg<<<(NNODES + 255) / 256, 256, 0, stream>>>(dinv);
  count_deg<<<(NEDGES + 255) / 256, 256, 0, stream>>>(ei, dinv);
  make_dinv<<<(NNODES + 255) / 256, 256, 0, stream>>>(dinv);

  // 4: XW = x @ W_gcn
  gemm_xw<<<(NNODES + 127) / 128, 256, 0, stream>>>(x, W_gcn, XW);

  // 5-7: spectral branch small matrices
  zero_buf<<<(KEIG * CCH + 255) / 256, 256, 0, stream>>>(Xhat, KEIG * CCH);
  project_vk<<<(NNODES + 1023) / 1024, 256, 0, stream>>>(Vk, x, Xhat);
  mix_spectral<<<1, 256, 0, stream>>>(Xhat, gain, W_lin, Y2);

  // 8: out = Vk@Y2 + XW*dinv^2 + b
  lift_base<<<(NNODES + 127) / 128, 256, 0, stream>>>(Vk, Y2, XW, dinv, b_gcn, out);

  // 9: edge aggregation
  edge_scatter<<<(NEDGES + 7) / 8, 256, 0, stream>>>(ei, XW, dinv, out);

  // 10: relu
  relu_k<<<(NNODES * CCH / 4 + 255) / 256, 256, 0, stream>>>(out);
}